// CalcRayColor_16604343566645
// MI455X (gfx1250) — compile-verified
//
#include <hip/hip_runtime.h>

// NeRF volume-render compositing for MI455X (gfx1250, wave32).
// One wave = 16 rays. Log-domain transmittance: cumprod(x) = exp(cumsum(-d*t)),
// cumsum done as a triangular f32 GEMM with V_WMMA_F32_16X16X4_F32:
//   - per 16-sample tile: 4 chained WMMAs over the diagonal band,
//   - inter-tile carry propagated through the WMMA C operand (row-replicated),
//     refreshed with a ds_swizzle broadcast of column N=15.
// Everything else (weights, 5 weighted reductions, weight store) fused in.
//
// ds_swizzle BITMASK_PERM encoding (probe-verified via disassembly round 1):
//   offset[15]=0, offset[14:10]=xor_mask, offset[9:5]=or_mask, offset[4:0]=and_mask
//   src_lane = ((lane & and) | or) ^ xor

typedef float v8f __attribute__((ext_vector_type(8)));
typedef float v2f __attribute__((ext_vector_type(2)));

#define NP  128
#define PAD 132   // LDS floats per ray: rows 16B-aligned; bank = (4*row + q) % 64 -> conflict-free

// broadcast lane 15 within each 16-lane half: src = (l & 0x10) | 0x0F
#define SWZ_BCAST15 0x01F0
// xor-swap by m within the wave (stays inside each 16-lane half for m < 16):
// src = (l & 0x1F) ^ m
#define SWZ_XOR(m) (((m) << 10) | 0x001F)

__device__ __forceinline__ float hsum16(float v) {
    // sum across the 16 lanes of each half-wave
    v += __int_as_float(__builtin_amdgcn_ds_swizzle(__float_as_int(v), SWZ_XOR(1)));
    v += __int_as_float(__builtin_amdgcn_ds_swizzle(__float_as_int(v), SWZ_XOR(2)));
    v += __int_as_float(__builtin_amdgcn_ds_swizzle(__float_as_int(v), SWZ_XOR(4)));
    v += __int_as_float(__builtin_amdgcn_ds_swizzle(__float_as_int(v), SWZ_XOR(8)));
    return v;
}

__global__ __launch_bounds__(256) void nerf_vol_render(
    const float* __restrict__ rgb,      // [3, NR, NP]
    const float* __restrict__ density,  // [NR, NP]
    const float* __restrict__ dists,    // [NR, NP]
    const float* __restrict__ zvals,    // [NR, NP]
    float* __restrict__ out,            // rgb(3*NR) | bg(NR) | depth(NR) | weight(NR*NP)
    int nr)
{
    __shared__ float lds_l[8 * 16 * PAD];   // 67.5 KB: per-wave private staging

    const int lane  = threadIdx.x & 31;
    const int wave  = threadIdx.x >> 5;
    const int group = blockIdx.x * 8 + wave;   // 16-ray group
    const int ray0  = group * 16;
    const size_t slab = (size_t)ray0 * NP;
    float* myLds = lds_l + wave * 16 * PAD;

    // ---- Phase 1: coalesced loads; l = -density*dists staged into padded LDS rows.
    // (wave-private region; DS ops from one wave complete in order -> no barrier)
    #pragma unroll 4
    for (int i = 0; i < 16; ++i) {
        const int q = lane * 4;
        const float4 d4 = *reinterpret_cast<const float4*>(density + slab + (size_t)i * NP + q);
        const float4 t4 = *reinterpret_cast<const float4*>(dists   + slab + (size_t)i * NP + q);
        float4 l4;
        l4.x = -d4.x * t4.x;
        l4.y = -d4.y * t4.y;
        l4.z = -d4.z * t4.z;
        l4.w = -d4.w * t4.w;
        *reinterpret_cast<float4*>(myLds + i * PAD + q) = l4;
    }

    const int laneN = lane & 15;         // column (N) within 16-lane half
    const int koff  = (lane >> 4) * 2;   // K offset carried by the lane half (A/B layout)
    const int rowA  = lane & 15;         // A-matrix row = ray within group

    // Band B tiles (constants): Bj[kq][np] = 1 iff local_q (= 4j+kq) <= np  (inclusive cumsum)
    v2f Bt[4];
    #pragma unroll
    for (int j = 0; j < 4; ++j) {
        Bt[j].x = (4 * j + koff + 0) <= laneN ? 1.0f : 0.0f;
        Bt[j].y = (4 * j + koff + 1) <= laneN ? 1.0f : 0.0f;
    }

    v8f C = {};                          // carry: inclusive log-sum at tile entry, row-replicated
    float aR[8], aG[8], aB[8], aZ[8], aW[8];
    #pragma unroll
    for (int j = 0; j < 8; ++j) { aR[j] = aG[j] = aB[j] = aZ[j] = aW[j] = 0.0f; }

    const int prevLane  = (lane & 16) | ((lane - 1) & 15);
    const size_t plane  = (size_t)nr * NP;
    const size_t wbase  = 5 * (size_t)nr;

    for (int n = 0; n < 8; ++n) {
        // D = carry + inclusive band-cumsum of this 16-sample tile (4 chained f32 WMMAs)
        v8f D = C;
        #pragma unroll
        for (int j = 0; j < 4; ++j) {
            const int q = 16 * n + 4 * j + koff;           // even -> 8B-aligned DS read
            const v2f A = *reinterpret_cast<const v2f*>(myLds + rowA * PAD + q);
            D = __builtin_amdgcn_wmma_f32_16x16x4_f32(false, A, false, Bt[j],
                                                      (short)0, D, false, false);
        }

        // next carry: broadcast N=15 value within each half (lane15 -> 0..15, lane31 -> 16..31)
        v8f Cn;
        #pragma unroll
        for (int j = 0; j < 8; ++j)
            Cn[j] = __int_as_float(__builtin_amdgcn_ds_swizzle(__float_as_int(D[j]), SWZ_BCAST15));

        // weight[p] = trans[p] - trans[p+1] = exp(incl[p-1]) - exp(incl[p])
        float w[8];
        #pragma unroll
        for (int j = 0; j < 8; ++j) {
            const float e = __expf(D[j]);
            float tp = __shfl(e, prevLane, 32);
            if (laneN == 0) tp = __expf(C[j]);             // trans at tile entry
            w[j] = tp - e;
        }
        C = Cn;

        // fused weighted sums + weight store (D-layout addressing)
        #pragma unroll
        for (int j = 0; j < 8; ++j) {
            const int ray = ray0 + j + ((lane >> 4) << 3);
            const int p   = 16 * n + laneN;
            const size_t off = (size_t)ray * NP + p;
            const float wj = w[j];
            aR[j] += wj * rgb[off];
            aG[j] += wj * rgb[plane + off];
            aB[j] += wj * rgb[2 * plane + off];
            aZ[j] += wj * zvals[off];
            aW[j] += wj;
            out[wbase + off] = wj;
        }
    }

    // ---- per-ray totals: reduce across the 16 lanes of each half, lane N==0 writes
    #pragma unroll
    for (int j = 0; j < 8; ++j) {
        const float sR = hsum16(aR[j]);
        const float sG = hsum16(aG[j]);
        const float sB = hsum16(aB[j]);
        const float sZ = hsum16(aZ[j]);
        const float sW = hsum16(aW[j]);
        if (laneN == 0) {
            const int ray = ray0 + j + ((lane >> 4) << 3);
            out[0 * (size_t)nr + ray] = sR;
            out[1 * (size_t)nr + ray] = sG;
            out[2 * (size_t)nr + ray] = sB;
            out[3 * (size_t)nr + ray] = 1.0f - sW;   // bg_alpha
            out[4 * (size_t)nr + ray] = sZ;          // depth
        }
    }
}

extern "C" void kernel_launch(void* const* d_in, const int* in_sizes, int n_in,
                              void* d_out, int out_size, void* d_ws, size_t ws_size,
                              hipStream_t stream) {
    // setup_inputs order: fg_vps (unused), batch_rgb, batch_density, batch_dists, batch_z_vals
    const float* rgb     = (const float*)d_in[1];
    const float* density = (const float*)d_in[2];
    const float* dists   = (const float*)d_in[3];
    const float* zvals   = (const float*)d_in[4];
    float* out = (float*)d_out;

    const int nr = in_sizes[4] / NP;        // 65536
    const int groups = nr / 16;             // 4096 sixteen-ray groups
    dim3 grid(groups / 8);                  // 8 waves (groups) per 256-thread block
    dim3 block(256);
    hipLaunchKernelGGL(nerf_vol_render, grid, block, 0, stream,
                       rgb, density, dists, zvals, out, nr);
}